// PlainKDEGrid_30288109372271
// MI455X (gfx1250) — compile-verified
//
#include <hip/hip_runtime.h>

#define GRID_H 31
#define GRID_W 51
#define PADR   30
#define KSIZE  61
#define NBATCH 16
#define NBOX   20
#define IN_H   1080
#define IN_W   1920
#define KPX    112   // 111 padded to multiple of 4
#define KPY    92    // 91 padded to multiple of 4
#define FPAD   64    // filtpad index offset: filtpad[FPAD+d] = filt[d] or 0
#define FPLEN  176   // covers d in [-63, 111]

typedef __attribute__((ext_vector_type(2))) float v2f;
typedef __attribute__((ext_vector_type(4))) float v4f;
typedef __attribute__((ext_vector_type(8))) float v8f;

// Banded-matrix convolution tile via V_WMMA_F32_16X16X4_F32.
//   out[16 x N] = In[16 x KP] * F[KP x N],  F[k][n] = filtpad[FPAD + k - n]
// A frag (16x4 f32, ISA 7.12.2): lanes 0-15 -> M=lane,   K={0,1}
//                                lanes 16-31 -> M=lane-16, K={2,3}
// B frag (4x16) mirrors A's K split; N = lane&15.
// C/D: VGPR r -> M=r (lanes 0-15), M=r+8 (lanes 16-31), N = lane&15.
template <int KP, int ISTRIDE, int OSTRIDE, int NREAL>
__device__ __forceinline__ void band_conv_wmma(
    const float* __restrict__ In, float* __restrict__ Out,
    const float* __restrict__ filtpad, int ntile, int lane)
{
    const int m    = lane & 15;
    const int koff = (lane >> 4) * 2;
    const int n    = ntile * 16 + (lane & 15);
    const float* arow = In + m * ISTRIDE + koff;          // 8B-aligned pairs
    const float* brow = filtpad + FPAD + koff - n;        // index >= 1 always

    v8f c = {};
    #pragma unroll
    for (int k0 = 0; k0 < KP; k0 += 4) {
        v2f a, bf;
        a.x  = arow[k0];
        a.y  = arow[k0 + 1];
        bf.x = brow[k0];
        bf.y = brow[k0 + 1];
        c = __builtin_amdgcn_wmma_f32_16x16x4_f32(
                false, a, false, bf, (short)0, c, false, false);
    }
    if (n < NREAL) {
        const int mhi = (lane >> 4) * 8;
        #pragma unroll
        for (int r = 0; r < 8; ++r)
            Out[(mhi + r) * OSTRIDE + n] = c[r];
    }
}

// ---------------------------------------------------------------------------
// Kernel 1: one workgroup computes xgrid[16][51], ygrid[16][31] into ws.
//   Stage A: separable KDE row/col sums (exact algebraic refactor of ref)
//   Stage B: 61-tap reflect conv as banded matmul (f32 WMMA, 12 waves)
//   Stage C: clip(o/w*2-1) -> ws
// ---------------------------------------------------------------------------
__global__ __launch_bounds__(1024) void kde_grid_kernel(
    const float* __restrict__ bboxes, float* __restrict__ ws)
{
    __shared__ float filtpad[FPLEN];
    __shared__ float Sxa[NBATCH*NBOX], Sya[NBATCH*NBOX];
    __shared__ float xacc[NBATCH][GRID_W], yacc[NBATCH][GRID_H];
    __shared__ float Tacc[NBATCH], Tfin[NBATCH];
    __shared__ float Xp[NBATCH][KPX], PXp[NBATCH][KPX];
    __shared__ float Yp[NBATCH][KPY], PYp[NBATCH][KPY];
    __shared__ float wxb[NBATCH][64], oxb[NBATCH][64];
    __shared__ float wyb[NBATCH][32], oyb[NBATCH][32];

    const int tid = threadIdx.x;

    // ---- stage 0: init accumulators + zero-padded gaussian filter taps ----
    for (int t = tid; t < NBATCH*GRID_W; t += 1024) xacc[t/GRID_W][t%GRID_W] = 0.f;
    for (int t = tid; t < NBATCH*GRID_H; t += 1024) yacc[t/GRID_H][t%GRID_H] = 0.f;
    if (tid < NBATCH) Tacc[tid] = 0.f;
    if (tid < FPLEN) {
        int d = tid - FPAD;
        float v = 0.f;
        if (d >= 0 && d < KSIZE) {
            float dd = (float)(d - PADR);
            // exp(-4*ln2*d^2 / FWHM^2), FWHM = 13
            v = expf(-(2.772588722239781f / 169.0f) * dd * dd);
        }
        filtpad[tid] = v;
    }
    __syncthreads();

    // ---- stage A0: per-(bbox,axis) Gaussian partition sums ----
    if (tid < NBATCH*NBOX*2) {
        int idx = tid >> 1, axis = tid & 1;
        int b = idx / NBOX, n = idx % NBOX;
        const float* bb = bboxes + (size_t)(b*NBOX + n)*4;
        if (axis == 0) {
            float cx = 0.5f*(bb[0]+bb[2]), w = bb[2]-bb[0];
            float s = 0.f;
            for (int i = 0; i < GRID_W; ++i) {
                float gx = (1920.0f * (float)i) / 50.0f;
                float dx = cx - gx;
                s += expf(-0.5f * dx*dx / w);
            }
            Sxa[idx] = s;
        } else {
            float cy = 0.5f*(bb[1]+bb[3]), h = bb[3]-bb[1];
            float s = 0.f;
            for (int j = 0; j < GRID_H; ++j) {
                float gy = (1080.0f * (float)j) / 30.0f;
                float dy = cy - gy;
                s += expf(-0.5f * dy*dy / h);
            }
            Sya[idx] = s;
        }
    }
    __syncthreads();

    // ---- stage A1: accumulate normalized separable contributions ----
    if (tid < NBATCH*NBOX*2) {
        int idx = tid >> 1, axis = tid & 1;
        int b = idx / NBOX, n = idx % NBOX;
        const float* bb = bboxes + (size_t)(b*NBOX + n)*4;
        float Sx = Sxa[idx], Sy = Sya[idx];
        float c = 1.0f / (1e-5f + Sx * Sy);
        if (axis == 0) {
            float cx = 0.5f*(bb[0]+bb[2]), w = bb[2]-bb[0];
            float scale = c * Sy;
            for (int i = 0; i < GRID_W; ++i) {
                float gx = (1920.0f * (float)i) / 50.0f;
                float dx = cx - gx;
                atomicAdd(&xacc[b][i], scale * expf(-0.5f * dx*dx / w));
            }
            atomicAdd(&Tacc[b], c * Sx * Sy);
        } else {
            float cy = 0.5f*(bb[1]+bb[3]), h = bb[3]-bb[1];
            float scale = c * Sx;
            for (int j = 0; j < GRID_H; ++j) {
                float gy = (1080.0f * (float)j) / 30.0f;
                float dy = cy - gy;
                atomicAdd(&yacc[b][j], scale * expf(-0.5f * dy*dy / h));
            }
        }
    }
    __syncthreads();

    if (tid < NBATCH) Tfin[tid] = Tacc[tid] + 31620.0f/3721.0f; // + 1581*20*bias
    __syncthreads();

    // ---- stage A2: build reflect-padded operands (and P*pad) in LDS ----
    for (int t = tid; t < NBATCH*KPX; t += 1024) {
        int b = t / KPX, k = t % KPX;
        float v = 0.f;
        if (k < GRID_W + 2*PADR) {                 // k < 111
            int s = k - PADR;
            if (s < 0) s = -s;
            if (s > GRID_W-1) s = 2*(GRID_W-1) - s;
            v = (xacc[b][s] + 620.0f/3721.0f) / Tfin[b];   // + 31*20*bias, /T
        }
        Xp[b][k]  = v;
        PXp[b][k] = v * ((float)(k - PADR) / (float)(GRID_W - 1));
    }
    for (int t = tid; t < NBATCH*KPY; t += 1024) {
        int b = t / KPY, k = t % KPY;
        float v = 0.f;
        if (k < GRID_H + 2*PADR) {                 // k < 91
            int s = k - PADR;
            if (s < 0) s = -s;
            if (s > GRID_H-1) s = 2*(GRID_H-1) - s;
            v = (yacc[b][s] + 1020.0f/3721.0f) / Tfin[b];  // + 51*20*bias, /T
        }
        Yp[b][k]  = v;
        PYp[b][k] = v * ((float)(k - PADR) / (float)(GRID_H - 1));
    }
    __syncthreads();

    // ---- stage B: banded-matrix conv via f32 WMMA (wave-uniform branches) ----
    {
        const int wave = tid >> 5;
        const int lane = tid & 31;
        if (wave < 8) {
            const float* In = (wave < 4) ? &Xp[0][0]  : &PXp[0][0];
            float*      Out = (wave < 4) ? &wxb[0][0] : &oxb[0][0];
            band_conv_wmma<KPX, KPX, 64, GRID_W>(In, Out, filtpad, wave & 3, lane);
        } else if (wave < 12) {
            const float* In = (wave < 10) ? &Yp[0][0]  : &PYp[0][0];
            float*      Out = (wave < 10) ? &wyb[0][0] : &oyb[0][0];
            band_conv_wmma<KPY, KPY, 32, GRID_H>(In, Out, filtpad, wave & 1, lane);
        }
    }
    __syncthreads();

    // ---- stage C: grids = clip(o/w * 2 - 1) -> ws ----
    for (int t = tid; t < NBATCH*GRID_W; t += 1024) {
        int b = t / GRID_W, i = t % GRID_W;
        float g = oxb[b][i] / wxb[b][i] * 2.0f - 1.0f;
        ws[b*GRID_W + i] = fminf(fmaxf(g, -1.0f), 1.0f);
    }
    for (int t = tid; t < NBATCH*GRID_H; t += 1024) {
        int b = t / GRID_H, j = t % GRID_H;
        float g = oyb[b][j] / wyb[b][j] * 2.0f - 1.0f;
        ws[NBATCH*GRID_W + b*GRID_H + j] = fminf(fmaxf(g, -1.0f), 1.0f);
    }
}

// ---------------------------------------------------------------------------
// Kernel 2: streaming writer (265 MB, HBM-store bound; NT stores since the
// write-once stream exceeds the 192 MB L2). Bilinear resize degenerates to
// two independent 1-D lerps. Each thread emits one aligned 16B NT store
// = {vx, vy, vx', vy} (two output pixels).
// ---------------------------------------------------------------------------
__global__ __launch_bounds__(256) void kde_resize_kernel(
    const float* __restrict__ ws, float* __restrict__ out)
{
    __shared__ float xg[GRID_W];
    __shared__ float yg[GRID_H];
    const int b = blockIdx.z;
    const int y = blockIdx.y;
    const int tid = threadIdx.x;

    if (tid < GRID_W) xg[tid] = ws[b*GRID_W + tid];
    if (tid < GRID_H) yg[tid] = ws[NBATCH*GRID_W + b*GRID_H + tid];
    __syncthreads();

    float sy = (float)y * ((float)(GRID_H - 1) / (float)(IN_H - 1));
    int y0 = (int)sy;
    int y1 = min(y0 + 1, GRID_H - 1);
    float ty = sy - (float)y0;
    float vy = yg[y0] * (1.0f - ty) + yg[y1] * ty;

    int x = blockIdx.x * 512 + tid * 2;
    if (x < IN_W) {
        float v0, v1;
        {
            float sx = (float)x * ((float)(GRID_W - 1) / (float)(IN_W - 1));
            int x0 = (int)sx; int x1i = min(x0 + 1, GRID_W - 1);
            float tx = sx - (float)x0;
            v0 = xg[x0] * (1.0f - tx) + xg[x1i] * tx;
        }
        {
            float sx = (float)(x + 1) * ((float)(GRID_W - 1) / (float)(IN_W - 1));
            int x0 = (int)sx; int x1i = min(x0 + 1, GRID_W - 1);
            float tx = sx - (float)x0;
            v1 = xg[x0] * (1.0f - tx) + xg[x1i] * tx;
        }
        v4f o = { v0, vy, v1, vy };
        v4f* dst = (v4f*)(out + ((size_t)(b * IN_H + y) * IN_W + x) * 2);
        __builtin_nontemporal_store(o, dst);
    }
}

extern "C" void kernel_launch(void* const* d_in, const int* in_sizes, int n_in,
                              void* d_out, int out_size, void* d_ws, size_t ws_size,
                              hipStream_t stream)
{
    // d_in[0] = imgs (unused: output depends only on batch size),
    // d_in[1] = gt_bboxes (16 x 20 x 4 float32)
    const float* bboxes = (const float*)d_in[1];
    float* ws = (float*)d_ws;   // xgrid[16][51] then ygrid[16][31]
    float* out = (float*)d_out; // (16, 1080, 1920, 2) float32

    kde_grid_kernel<<<1, 1024, 0, stream>>>(bboxes, ws);

    dim3 grid((IN_W + 511) / 512, IN_H, NBATCH); // (4, 1080, 16)
    kde_resize_kernel<<<grid, 256, 0, stream>>>(ws, out);
}